// SAGE_28398323761564
// MI455X (gfx1250) — compile-verified
//
#include <hip/hip_runtime.h>

#define N_NODES 40000
#define N_EDGES 640000
#define DIM     128

typedef __attribute__((ext_vector_type(2))) float v2f;
typedef __attribute__((ext_vector_type(4))) float v4f;
typedef __attribute__((ext_vector_type(8))) float v8f;

#if defined(__HIP_DEVICE_COMPILE__) && __has_builtin(__builtin_amdgcn_wmma_f32_16x16x4_f32)
#define HAVE_WMMA_F32 1
#endif

// ---------------------------------------------------------------- zero (float4)
__global__ void zero_f4(float* __restrict__ p, int n4) {
    int i = blockIdx.x * blockDim.x + threadIdx.x;
    if (i < n4) {
        v4f z = {};
        ((v4f*)p)[i] = z;
    }
}

// ---------------------------------------------------------------- degree accum
__global__ void deg_accum(const int* __restrict__ dst, float* __restrict__ deg) {
    int e = blockIdx.x * blockDim.x + threadIdx.x;
    if (e < N_EDGES) {
        __hip_atomic_fetch_add(&deg[dst[e]], 1.0f,
                               __ATOMIC_RELAXED, __HIP_MEMORY_SCOPE_AGENT);
    }
}

// ---------------------------------------------------------------- 1 / max(deg,1)
__global__ void deg_invert(float* __restrict__ deg) {
    int i = blockIdx.x * blockDim.x + threadIdx.x;
    if (i < N_NODES) {
        float d = deg[i];
        deg[i] = 1.0f / fmaxf(d, 1.0f);
    }
}

// ---------------------------------------------------------------- edge scatter
// one wave (32 lanes) per edge: lane loads float4 of the source row (coalesced
// 512B read) and atomically adds into the destination row.
__global__ void edge_scatter(const float* __restrict__ h,
                             const int* __restrict__ src,
                             const int* __restrict__ dst,
                             float* __restrict__ agg) {
    int t    = blockIdx.x * blockDim.x + threadIdx.x;
    int e    = t >> 5;
    int lane = t & 31;
    if (e >= N_EDGES) return;

    int s = __builtin_amdgcn_readfirstlane(src[e]);
    int d = __builtin_amdgcn_readfirstlane(dst[e]);

    const float* hp = h + (size_t)s * DIM + lane * 4;
    float*       ap = agg + (size_t)d * DIM + lane * 4;
    v4f v = *(const v4f*)hp;
    __hip_atomic_fetch_add(ap + 0, v.x, __ATOMIC_RELAXED, __HIP_MEMORY_SCOPE_AGENT);
    __hip_atomic_fetch_add(ap + 1, v.y, __ATOMIC_RELAXED, __HIP_MEMORY_SCOPE_AGENT);
    __hip_atomic_fetch_add(ap + 2, v.z, __ATOMIC_RELAXED, __HIP_MEMORY_SCOPE_AGENT);
    __hip_atomic_fetch_add(ap + 3, v.w, __ATOMIC_RELAXED, __HIP_MEMORY_SCOPE_AGENT);
}

// ---------------------------------------------------------------- fused SAGE layer GEMM
// out[N,128] = h @ Wself + (agg * inv_deg) @ Wneigh + b  (+ optional ReLU)
// block = 256 threads = 8 waves; block -> 16-row M tile; wave -> one 16-col N tile.
// K loop: 32 steps of K=4, two V_WMMA_F32_16X16X4_F32 per step into one v8f C.
__global__ void sage_gemm(const float* __restrict__ h,
                          const float* __restrict__ agg,
                          const float* __restrict__ invdeg,
                          const float* __restrict__ wself,
                          const float* __restrict__ wneigh,
                          const float* __restrict__ bias,
                          float* __restrict__ out,
                          int do_relu) {
    const int lane  = threadIdx.x & 31;
    const int wv    = threadIdx.x >> 5;        // 0..7 : N tile
    const int m     = lane & 15;               // A-matrix row (M) for this lane
    const int khalf = (lane >> 4) << 1;        // lanes 16-31 carry K+2,K+3
    const int row   = blockIdx.x * 16 + m;
    const int col   = wv * 16 + (lane & 15);   // B-matrix column (N) for this lane

    const float invd = invdeg[row];
    const float* hrow = h   + (size_t)row * DIM;
    const float* grow = agg + (size_t)row * DIM;

    v8f acc = {};
#if HAVE_WMMA_F32
#pragma unroll 8
    for (int k = 0; k < DIM; k += 4) {
        const int kk = k + khalf;
        v2f a_h = *(const v2f*)(hrow + kk);
        v2f a_g = (*(const v2f*)(grow + kk)) * invd;   // fold mean normalization
        v2f b_s = { wself [kk * DIM + col], wself [(kk + 1) * DIM + col] };
        v2f b_n = { wneigh[kk * DIM + col], wneigh[(kk + 1) * DIM + col] };
        acc = __builtin_amdgcn_wmma_f32_16x16x4_f32(false, a_h, false, b_s,
                                                    (short)0, acc, false, false);
        acc = __builtin_amdgcn_wmma_f32_16x16x4_f32(false, a_g, false, b_n,
                                                    (short)0, acc, false, false);
    }
#else
    // scalar fallback (host-pass parse / builtin missing): same C layout
    {
        const int n0     = lane & 15;
        const int outcol = wv * 16 + n0;
        const int mbase  = blockIdx.x * 16 + ((lane >> 4) << 3);
        for (int r = 0; r < 8; ++r) {
            float s = 0.0f;
            const float* hr = h   + (size_t)(mbase + r) * DIM;
            const float* gr = agg + (size_t)(mbase + r) * DIM;
            float id = invdeg[mbase + r];
            for (int k = 0; k < DIM; ++k)
                s += hr[k] * wself[k * DIM + outcol] + id * gr[k] * wneigh[k * DIM + outcol];
            acc[r] = s;
        }
    }
#endif

    // C/D layout: VGPR r -> M = r (lanes 0-15) or M = r+8 (lanes 16-31), N = lane%16
    const int n0     = lane & 15;
    const int outcol = wv * 16 + n0;
    const int mbase  = blockIdx.x * 16 + ((lane >> 4) << 3);
    const float bv   = bias[outcol];
#pragma unroll
    for (int r = 0; r < 8; ++r) {
        float v = acc[r] + bv;
        if (do_relu) v = fmaxf(v, 0.0f);
        out[(size_t)(mbase + r) * DIM + outcol] = v;
    }
}

// ---------------------------------------------------------------- launcher
extern "C" void kernel_launch(void* const* d_in, const int* in_sizes, int n_in,
                              void* d_out, int out_size, void* d_ws, size_t ws_size,
                              hipStream_t stream) {
    const float* x      = (const float*)d_in[0];
    const int*   src    = (const int*)  d_in[1];
    const int*   dst    = (const int*)  d_in[2];
    const float* wself  = (const float*)d_in[3];
    const float* wneigh = (const float*)d_in[4];
    const float* b      = (const float*)d_in[5];
    float*       out    = (float*)d_out;

    // workspace layout (floats): invdeg | h0 | h1 | agg  (~59 MiB total)
    float* ws     = (float*)d_ws;
    float* invdeg = ws;
    float* h0     = ws + 40960;                       // N_NODES rounded up
    float* h1     = h0 + (size_t)N_NODES * DIM;
    float* agg    = h1 + (size_t)N_NODES * DIM;

    // degree -> inv_deg
    zero_f4  <<<(N_NODES / 4 + 255) / 256, 256, 0, stream>>>(invdeg, N_NODES / 4);
    deg_accum<<<(N_EDGES + 255) / 256,     256, 0, stream>>>(dst, invdeg);
    deg_invert<<<(N_NODES + 255) / 256,    256, 0, stream>>>(invdeg);

    const float* hin = x;
    float* houts[3] = { h0, h1, out };
    const int nd4 = N_NODES * DIM / 4;
    for (int l = 0; l < 3; ++l) {
        zero_f4<<<(nd4 + 255) / 256, 256, 0, stream>>>(agg, nd4);
        edge_scatter<<<N_EDGES / 8, 256, 0, stream>>>(hin, src, dst, agg);
        sage_gemm<<<N_NODES / 16, 256, 0, stream>>>(
            hin, agg, invdeg,
            wself + (size_t)l * DIM * DIM,
            wneigh + (size_t)l * DIM * DIM,
            b + (size_t)l * DIM,
            houts[l], (l < 2) ? 1 : 0);
        hin = houts[l];
    }
}